// EKF_80753975099560
// MI455X (gfx1250) — compile-verified
//
#include <hip/hip_runtime.h>

// ---------------------------------------------------------------------------
// Batched EKF update for MI455X (gfx1250, wave32).
// One wave = one batch. All 16x16 matrix products on V_WMMA_F32_16X16X4_F32.
// Memory-bound problem (~193MB @ 23.3TB/s ~ 8us floor): fp32 throughout,
// coalesced float4 staging into per-wave LDS, WMMA for all dense products.
// ---------------------------------------------------------------------------

typedef __attribute__((ext_vector_type(2))) float v2f;
typedef __attribute__((ext_vector_type(4))) float v4f;
typedef __attribute__((ext_vector_type(8))) float v8f;

#define BATCH 32768
#define WAVES_PER_BLOCK 4
#define THREADS (WAVES_PER_BLOCK * 32)

struct alignas(16) WaveLDS {
    float buf[5][256];   // five 16x16 f32 tiles, heavily reused
    float aug[128];      // 8x16 augmented [S | I] for Gauss-Jordan
    float vec[32];       // [0:16)=x_pri, [16:24)=e
};

__device__ __forceinline__ v8f wmma4(v2f a, v2f b, v8f c) {
    // D = A(16x4,f32) x B(4x16,f32) + C(16x16,f32)
    return __builtin_amdgcn_wmma_f32_16x16x4_f32(
        false, a, false, b, (short)0, c, false, false);
}

// acc += A(16x16 row-major LDS) @ B(16x16 row-major LDS)
__device__ __forceinline__ v8f mm_AB(const float* A, const float* B, v8f acc, int lane) {
    const int lh = lane >> 4, lc = lane & 15;
#pragma unroll
    for (int t = 0; t < 4; ++t) {
        const int k0 = 4 * t + 2 * lh;
        v2f a, b;
        a.x = A[lc * 16 + k0];       a.y = A[lc * 16 + k0 + 1];
        b.x = B[k0 * 16 + lc];       b.y = B[(k0 + 1) * 16 + lc];
        acc = wmma4(a, b, acc);
    }
    return acc;
}

// acc += A(16x16 row-major LDS) @ B^T  (row-per-lane load of B == B^T in B-layout)
__device__ __forceinline__ v8f mm_ABt(const float* A, const float* Bt, v8f acc, int lane) {
    const int lh = lane >> 4, lc = lane & 15;
#pragma unroll
    for (int t = 0; t < 4; ++t) {
        const int k0 = 4 * t + 2 * lh;
        v2f a, b;
        a.x = A[lc * 16 + k0];       a.y = A[lc * 16 + k0 + 1];
        b.x = Bt[lc * 16 + k0];      b.y = Bt[lc * 16 + k0 + 1];
        acc = wmma4(a, b, acc);
    }
    return acc;
}

// store D-layout accumulator (rows i+8*lh, col lane&15) to row-major LDS
__device__ __forceinline__ void st_D(float* Mo, v8f d, int lane) {
    const int lh = lane >> 4, lc = lane & 15;
#pragma unroll
    for (int i = 0; i < 8; ++i) Mo[(i + 8 * lh) * 16 + lc] = d[i];
}

// load row-major LDS matrix into D-layout accumulator
__device__ __forceinline__ v8f ld_D(const float* Mi, int lane) {
    const int lh = lane >> 4, lc = lane & 15;
    v8f d;
#pragma unroll
    for (int i = 0; i < 8; ++i) d[i] = Mi[(i + 8 * lh) * 16 + lc];
    return d;
}

__global__ __launch_bounds__(THREADS) void ekf_kernel(
    const float* __restrict__ gx, const float* __restrict__ gy,
    const float* __restrict__ gu, const float* __restrict__ gP,
    const float* __restrict__ gA, const float* __restrict__ gB,
    const float* __restrict__ gC, const float* __restrict__ gD,
    const float* __restrict__ gc1, const float* __restrict__ gc2,
    const float* __restrict__ gQ, const float* __restrict__ gR,
    float* __restrict__ out_x, float* __restrict__ out_P)
{
    __shared__ WaveLDS lds[WAVES_PER_BLOCK];

    const int lane = threadIdx.x & 31;
    const int wave = threadIdx.x >> 5;
    const int b = blockIdx.x * WAVES_PER_BLOCK + wave;
    if (b >= BATCH) return;                 // wave-uniform

    const int lh = lane >> 4, lc = lane & 15;
    WaveLDS& W = lds[wave];
    float* mA    = W.buf[0];   // A, later S^-1 (padded 16x16)
    float* mB1   = W.buf[1];   // P -> CP -> P*C^T -> (-K*C)
    float* mB2   = W.buf[2];   // AP -> K (padded 16x16)
    float* mPpri = W.buf[3];   // P_pri (live to the end)
    float* mC    = W.buf[4];   // C zero-padded to 16x16
    float* aug   = W.aug;

    // ---- stage A, P, Cpad into LDS (coalesced float4) --------------------
    {
        const v4f* A4 = (const v4f*)(gA + (size_t)b * 256);
        const v4f* P4 = (const v4f*)(gP + (size_t)b * 256);
        const v4f* C4 = (const v4f*)(gC + (size_t)b * 128);
        v4f* mA4 = (v4f*)mA; v4f* mP4 = (v4f*)mB1; v4f* mC4 = (v4f*)mC;
#pragma unroll
        for (int i = 0; i < 2; ++i) mA4[lane + 32 * i] = A4[lane + 32 * i];
#pragma unroll
        for (int i = 0; i < 2; ++i) mP4[lane + 32 * i] = P4[lane + 32 * i];
#pragma unroll
        for (int i = 0; i < 8; ++i) mC[lane + 32 * i] = 0.0f;  // zero 16x16
        mC4[lane] = C4[lane];                                   // rows 0..7 = C
    }
    __builtin_amdgcn_wave_barrier();

    // ---- x_pri = A x + B u + c1  (lanes 0..15, VALU) ---------------------
    if (lane < 16) {
        float s = gc1[(size_t)b * 16 + lane];
#pragma unroll
        for (int j = 0; j < 16; ++j) s += mA[lane * 16 + j] * gx[(size_t)b * 16 + j];
#pragma unroll
        for (int j = 0; j < 8; ++j)  s += gB[(size_t)b * 128 + lane * 8 + j] * gu[(size_t)b * 8 + j];
        W.vec[lane] = s;
    }

    const v8f zero = {};

    // ---- 1) AP = A @ P ---------------------------------------------------
    v8f acc = mm_AB(mA, mB1, zero, lane);
    st_D(mB2, acc, lane);                                   // buf2 = AP
    __builtin_amdgcn_wave_barrier();

    // ---- 2) P_pri = AP @ A^T + Q  (Q loaded straight into accumulator) ---
    v8f q;
#pragma unroll
    for (int i = 0; i < 8; ++i)
        q[i] = gQ[(size_t)b * 256 + (i + 8 * lh) * 16 + lc];
    acc = mm_ABt(mB2, mA, q, lane);
    st_D(mPpri, acc, lane);                                 // buf3 = P_pri
    __builtin_amdgcn_wave_barrier();

    // ---- 3) CP = Cpad @ P_pri  (rows 8..15 exactly zero) -----------------
    acc = mm_AB(mC, mPpri, zero, lane);
    st_D(mB1, acc, lane);                                   // buf1 = CP
    __builtin_amdgcn_wave_barrier();

    // ---- 4) S = CP @ C^T + Rpad  (top-left 8x8 is S) ---------------------
    v8f racc = {};
    if (lh == 0 && lc < 8) {
#pragma unroll
        for (int i = 0; i < 8; ++i)
            racc[i] = gR[(size_t)b * 64 + i * 8 + lc];
    }
    acc = mm_ABt(mB1, mC, racc, lane);

    // scatter S (8x8) into augmented [S | I]
    if (lh == 0 && lc < 8) {
#pragma unroll
        for (int i = 0; i < 8; ++i) aug[i * 16 + lc] = acc[i];
    }
#pragma unroll
    for (int j = 0; j < 2; ++j) {                           // identity half
        int e2 = lane + 32 * j;                             // 0..63
        int rr = e2 >> 3, cc = e2 & 7;
        aug[rr * 16 + 8 + cc] = (rr == cc) ? 1.0f : 0.0f;
    }
    __builtin_amdgcn_wave_barrier();

    // ---- 5) Gauss-Jordan inverse of SPD S (pivot-free) -------------------
    for (int k = 0; k < 8; ++k) {
        float pinv = 1.0f / aug[k * 16 + k];
        if (lane < 16) aug[k * 16 + lane] *= pinv;          // scale pivot row
        __builtin_amdgcn_wave_barrier();
        float rowk = aug[k * 16 + lc];
        float f[4];
#pragma unroll
        for (int j = 0; j < 4; ++j) f[j] = aug[(lh + 2 * j) * 16 + k];
        __builtin_amdgcn_wave_barrier();
#pragma unroll
        for (int j = 0; j < 4; ++j) {
            int rr = lh + 2 * j;
            if (rr != k) aug[rr * 16 + lc] -= f[j] * rowk;
        }
        __builtin_amdgcn_wave_barrier();
    }
    // buf0 (A is dead): S^-1 zero-padded to 16x16
#pragma unroll
    for (int i = 0; i < 8; ++i) mA[lane + 32 * i] = 0.0f;
#pragma unroll
    for (int j = 0; j < 2; ++j) {
        int e2 = lane + 32 * j;
        int rr = e2 >> 3, cc = e2 & 7;
        mA[rr * 16 + cc] = aug[rr * 16 + 8 + cc];
    }
    __builtin_amdgcn_wave_barrier();

    // ---- 6) PCt = P_pri @ C^T  (cols 8..15 exactly zero) -----------------
    acc = mm_ABt(mPpri, mC, zero, lane);
    st_D(mB1, acc, lane);                                   // buf1 = P*C^T
    __builtin_amdgcn_wave_barrier();

    // ---- 7) K = PCt @ Sinv_pad  (cols 8..15 exactly zero) ----------------
    acc = mm_AB(mB1, mA, zero, lane);
    st_D(mB2, acc, lane);                                   // buf2 = K
    __builtin_amdgcn_wave_barrier();

    // ---- 8) KC = K @ Cpad, stored negated --------------------------------
    acc = mm_AB(mB2, mC, zero, lane);
#pragma unroll
    for (int i = 0; i < 8; ++i)
        mB1[(i + 8 * lh) * 16 + lc] = -acc[i];              // buf1 = -K*C
    __builtin_amdgcn_wave_barrier();

    // ---- e = y - C x_pri - D u - c2  (lanes 0..7) ------------------------
    if (lane < 8) {
        float s = gy[(size_t)b * 8 + lane] - gc2[(size_t)b * 8 + lane];
#pragma unroll
        for (int j = 0; j < 16; ++j) s -= mC[lane * 16 + j] * W.vec[j];
#pragma unroll
        for (int j = 0; j < 8; ++j)  s -= gD[(size_t)b * 64 + lane * 8 + j] * gu[(size_t)b * 8 + j];
        W.vec[16 + lane] = s;
    }
    __builtin_amdgcn_wave_barrier();

    // ---- 9) P_post = (-KC) @ P_pri + P_pri  -> global --------------------
    v8f pp = ld_D(mPpri, lane);
    acc = mm_AB(mB1, mPpri, pp, lane);
    {
        float* oP = out_P + (size_t)b * 256;
#pragma unroll
        for (int i = 0; i < 8; ++i)
            oP[(i + 8 * lh) * 16 + lc] = acc[i];
    }

    // ---- x_post = x_pri + K e  (lanes 0..15) -> global -------------------
    if (lane < 16) {
        float s = W.vec[lane];
#pragma unroll
        for (int j = 0; j < 8; ++j) s += mB2[lane * 16 + j] * W.vec[16 + j];
        out_x[(size_t)b * 16 + lane] = s;
    }
}

extern "C" void kernel_launch(void* const* d_in, const int* in_sizes, int n_in,
                              void* d_out, int out_size, void* d_ws, size_t ws_size,
                              hipStream_t stream) {
    (void)in_sizes; (void)n_in; (void)out_size; (void)d_ws; (void)ws_size;
    // setup_inputs order: x, y, u, P, A, B, C, D, c1, c2, Q, R
    const float* gx  = (const float*)d_in[0];
    const float* gy  = (const float*)d_in[1];
    const float* gu  = (const float*)d_in[2];
    const float* gP  = (const float*)d_in[3];
    const float* gA  = (const float*)d_in[4];
    const float* gB  = (const float*)d_in[5];
    const float* gC  = (const float*)d_in[6];
    const float* gD  = (const float*)d_in[7];
    const float* gc1 = (const float*)d_in[8];
    const float* gc2 = (const float*)d_in[9];
    const float* gQ  = (const float*)d_in[10];
    const float* gR  = (const float*)d_in[11];

    float* out_x = (float*)d_out;                    // [B,16]
    float* out_P = out_x + (size_t)BATCH * 16;       // [B,16,16]

    dim3 grid(BATCH / WAVES_PER_BLOCK);
    dim3 block(THREADS);
    ekf_kernel<<<grid, block, 0, stream>>>(gx, gy, gu, gP, gA, gB, gC, gD,
                                           gc1, gc2, gQ, gR, out_x, out_P);
}